// LRTModel_45011257262754
// MI455X (gfx1250) — compile-verified
//
#include <hip/hip_runtime.h>
#include <math.h>

#define H      128
#define BATCH  128
#define NNODES 2047
#define AS_STRIDE 260
#define CS_STRIDE 260

typedef float v2f __attribute__((ext_vector_type(2)));
typedef float v8f __attribute__((ext_vector_type(8)));

__device__ __forceinline__ float sigm(float x) { return 1.0f / (1.0f + __expf(-x)); }

// ---------------------------------------------------------------------------
// Precompute T_iou[9][384] = emb@W_iou + b_iou ; T_f[9][128] = emb@W_f + b_f
// ---------------------------------------------------------------------------
__global__ void prep_tables_kernel(const float* __restrict__ emb,
                                   const float* __restrict__ Wiou,
                                   const float* __restrict__ biou,
                                   const float* __restrict__ Wf,
                                   const float* __restrict__ bf,
                                   float* __restrict__ Tiou,
                                   float* __restrict__ Tf) {
  int idx = blockIdx.x * 256 + threadIdx.x;
  if (idx < 9 * 384) {
    int v = idx / 384, c = idx % 384;
    float s = biou[c];
    for (int i = 0; i < 128; ++i) s += emb[v * 128 + i] * Wiou[i * 384 + c];
    Tiou[idx] = s;
  } else if (idx < 9 * 384 + 9 * 128) {
    int q = idx - 9 * 384;
    int v = q / 128, c = q % 128;
    float s = bf[c];
    for (int i = 0; i < 128; ++i) s += emb[v * 128 + i] * Wf[i * 128 + c];
    Tf[q] = s;
  }
}

// ---------------------------------------------------------------------------
// Pack the 256x640 concatenated recurrent weight matrix
//   Wcat[rr][col]: rr<128 -> left-child row rr, rr>=128 -> right-child row
//   col<384: U_iou[child] ; 384..511: U_f[0][child] ; 512..639: U_f[1][child]
// into per-lane WMMA B-fragments (float2 per lane; B row = vgpr + 2*half).
// Layout: Wpack[((k4*40 + ct)*32 + lane)*2 + {0,1}]
// ---------------------------------------------------------------------------
__device__ __forceinline__ float wcat_elem(int rr, int col,
                                           const float* __restrict__ Uiou,
                                           const float* __restrict__ Uf) {
  int child = rr >> 7;
  int i = rr & 127;
  if (col < 384) return Uiou[((size_t)child * 128 + i) * 384 + col];
  int a = (col >= 512) ? 1 : 0;
  int jj = col & 127;  // 384 and 512 are multiples of 128
  return Uf[(((size_t)(a * 2 + child)) * 128 + i) * 128 + jj];
}

__global__ void pack_weights_kernel(const float* __restrict__ Uiou,
                                    const float* __restrict__ Uf,
                                    float* __restrict__ Wpack) {
  int idx = blockIdx.x * 256 + threadIdx.x;
  if (idx >= 64 * 40 * 32) return;
  int k4   = idx / 1280;
  int rem  = idx % 1280;
  int ct   = rem >> 5;
  int lane = rem & 31;
  int n    = lane & 15;
  int half = lane >> 4;
  int col  = ct * 16 + n;
  int rr   = k4 * 4 + 2 * half;
  Wpack[2 * idx]     = wcat_elem(rr,     col, Uiou, Uf);
  Wpack[2 * idx + 1] = wcat_elem(rr + 1, col, Uiou, Uf);
}

// ---------------------------------------------------------------------------
// One tree level: GEMM [ROWS x 256] @ [256 x 640] via V_WMMA_F32_16X16X4_F32,
// fused with the full LSTM cell in accumulator registers.
//   Level buffers are [M_nodes, BATCH, H]; child local index = 2p + side.
//   LEAF=true: children are leaves -> compute their (h,c) from T-tables.
// Wave w owns h-columns j = w*16 + n16 and accumulators (i,o,u,fl,fr) for
// those same columns (col tiles w, w+8, w+16, w+24, w+32).
// ---------------------------------------------------------------------------
template <bool LEAF, int RT>
__global__ __launch_bounds__(256) void tree_level_kernel(
    const float* __restrict__ childH, const float* __restrict__ childC,
    const int* __restrict__ tokens, int nodeBase,
    const float* __restrict__ Wpack, const float* __restrict__ Tiou,
    const float* __restrict__ Tf, float* __restrict__ outH,
    float* __restrict__ outC) {
  constexpr int ROWS = RT * 16;
  __shared__ float As[ROWS * AS_STRIDE];  // [row][0:128)=hl, [128:256)=hr
  __shared__ float Cs[ROWS * CS_STRIDE];  // [row][0:128)=cl, [128:256)=cr
  __shared__ int tokP[ROWS];

  const int rowBase = blockIdx.x * ROWS;  // ROWS <= 64 -> single parent p
  const int p = rowBase / BATCH;
  const int bBase = rowBase % BATCH;
  const int t = threadIdx.x;

  if (t < ROWS)
    tokP[t] = tokens[(size_t)(bBase + t) * NNODES + nodeBase + p];

  if (LEAF) {
    // Children are leaves: h = sig(o)*tanh(sig(i)*tanh(u)), c = sig(i)*tanh(u)
    for (int q = t; q < ROWS * 256; q += 256) {
      int m = q >> 8;
      int cc = q & 255;
      int side = cc >> 7;
      int j = cc & 127;
      int node = 2 * nodeBase + 1 + 2 * p + side;
      int tok = tokens[(size_t)(bBase + m) * NNODES + node];
      float iv = Tiou[tok * 384 + j];
      float ov = Tiou[tok * 384 + 128 + j];
      float uv = Tiou[tok * 384 + 256 + j];
      float ig = sigm(iv), og = sigm(ov), ug = tanhf(uv);
      float c = ig * ug;
      float h = og * tanhf(c);
      As[m * AS_STRIDE + cc] = h;
      Cs[m * CS_STRIDE + cc] = c;
    }
  } else {
    const size_t chBase = ((size_t)(2 * p) * BATCH + bBase) * H;
    const float4* hL = (const float4*)(childH + chBase);
    const float4* hR = (const float4*)(childH + chBase + (size_t)BATCH * H);
    const float4* cL = (const float4*)(childC + chBase);
    const float4* cR = (const float4*)(childC + chBase + (size_t)BATCH * H);
    for (int q = t; q < ROWS * 32; q += 256) {
      int m = q >> 5;
      int j4 = q & 31;
      float4 a = hL[m * 32 + j4];
      float4 b = hR[m * 32 + j4];
      float4 c = cL[m * 32 + j4];
      float4 d = cR[m * 32 + j4];
      int j = j4 << 2;
      *(float4*)&As[m * AS_STRIDE + j] = a;
      *(float4*)&As[m * AS_STRIDE + 128 + j] = b;
      *(float4*)&Cs[m * CS_STRIDE + j] = c;
      *(float4*)&Cs[m * CS_STRIDE + 128 + j] = d;
    }
  }
  __syncthreads();

  const int wave = t >> 5;
  const int lane = t & 31;
  const int n16 = lane & 15;
  const int half = lane >> 4;

  v8f zero = {};
  v8f acc[5][RT];
#pragma unroll
  for (int cc = 0; cc < 5; ++cc)
#pragma unroll
    for (int rt = 0; rt < RT; ++rt) acc[cc][rt] = zero;

  const float* wptr[5];
#pragma unroll
  for (int cc = 0; cc < 5; ++cc) {
    int ct = wave + 8 * cc;  // col tiles: i, o, u, fl, fr for columns wave*16+n
    wptr[cc] = Wpack + ((size_t)ct * 32 + lane) * 2;
  }

#pragma unroll 4
  for (int k4 = 0; k4 < 64; ++k4) {
    int k = k4 * 4;
    v2f a[RT];
#pragma unroll
    for (int rt = 0; rt < RT; ++rt)
      a[rt] = *(const v2f*)&As[(rt * 16 + n16) * AS_STRIDE + k + 2 * half];
#pragma unroll
    for (int cc = 0; cc < 5; ++cc) {
      v2f bf = *(const v2f*)(wptr[cc] + (size_t)k4 * 2560);
#pragma unroll
      for (int rt = 0; rt < RT; ++rt)
        acc[cc][rt] = __builtin_amdgcn_wmma_f32_16x16x4_f32(
            false, a[rt], false, bf, (short)0, acc[cc][rt], false, false);
    }
  }

  // LSTM cell entirely in accumulator registers
  const int j = (wave << 4) + n16;
#pragma unroll
  for (int rt = 0; rt < RT; ++rt) {
#pragma unroll
    for (int r = 0; r < 8; ++r) {
      int rowL = rt * 16 + r + 8 * half;  // C/D layout: M = r + 8*half
      int tok = tokP[rowL];
      float iv = acc[0][rt][r] + Tiou[tok * 384 + j];
      float ov = acc[1][rt][r] + Tiou[tok * 384 + 128 + j];
      float uv = acc[2][rt][r] + Tiou[tok * 384 + 256 + j];
      float tfb = Tf[tok * 128 + j];
      float flv = acc[3][rt][r] + tfb;
      float frv = acc[4][rt][r] + tfb;
      float cl = Cs[rowL * CS_STRIDE + j];
      float cr = Cs[rowL * CS_STRIDE + 128 + j];
      float ig = sigm(iv), og = sigm(ov), ug = tanhf(uv);
      float fl = sigm(flv), fr = sigm(frv);
      float cn = ig * ug + fl * cl + fr * cr;
      float hn = og * tanhf(cn);
      size_t o = ((size_t)(rowBase + rowL)) * H + j;
      outH[o] = hn;
      outC[o] = cn;
    }
  }
}

// ---------------------------------------------------------------------------
// out[n,k] = tanh( sum_ij A[i,j,k] hA[n,i] hB[n,j] + hA@U1 + hB@U2 + b_out )
// ---------------------------------------------------------------------------
__global__ __launch_bounds__(128) void combine_kernel(
    const float* __restrict__ Amat, const float* __restrict__ U1,
    const float* __restrict__ U2, const float* __restrict__ bout,
    const float* __restrict__ hA, const float* __restrict__ hB,
    float* __restrict__ out) {
  __shared__ float sA[128], sB[128], red[128 * 7];
  int n = blockIdx.x, t = threadIdx.x;
  sA[t] = hA[n * 128 + t];
  sB[t] = hB[n * 128 + t];
  __syncthreads();
  float wa = sA[t];
  float accv[7];
#pragma unroll
  for (int k = 0; k < 7; ++k)
    accv[k] = wa * U1[t * 7 + k] + sB[t] * U2[t * 7 + k];
  const float* ap = Amat + (size_t)t * 128 * 7;
  for (int jj = 0; jj < 128; ++jj) {
    float w = wa * sB[jj];
#pragma unroll
    for (int k = 0; k < 7; ++k) accv[k] += w * ap[jj * 7 + k];
  }
#pragma unroll
  for (int k = 0; k < 7; ++k) red[t * 7 + k] = accv[k];
  __syncthreads();
  if (t < 7) {
    float s = bout[t];
    for (int m = 0; m < 128; ++m) s += red[m * 7 + t];
    out[n * 7 + t] = tanhf(s);
  }
}

// ---------------------------------------------------------------------------
extern "C" void kernel_launch(void* const* d_in, const int* in_sizes, int n_in,
                              void* d_out, int out_size, void* d_ws,
                              size_t ws_size, hipStream_t stream) {
  (void)in_sizes; (void)n_in; (void)out_size; (void)ws_size;
  const int*   xa   = (const int*)d_in[0];
  const int*   xb   = (const int*)d_in[1];
  const float* emb  = (const float*)d_in[2];
  const float* Wiou = (const float*)d_in[3];
  const float* biou = (const float*)d_in[4];
  const float* Wf   = (const float*)d_in[5];
  const float* bf   = (const float*)d_in[6];
  const float* Uiou = (const float*)d_in[7];
  const float* Uf   = (const float*)d_in[8];
  const float* Amat = (const float*)d_in[9];
  const float* U1   = (const float*)d_in[10];
  const float* U2   = (const float*)d_in[11];
  const float* bout = (const float*)d_in[12];
  float* out = (float*)d_out;

  char* ws = (char*)d_ws;
  float* Tiou  = (float*)(ws + 0);        // 13,824 B
  float* Tf    = (float*)(ws + 16384);    //  4,608 B
  float* Wpack = (float*)(ws + 32768);    // 655,360 B
  size_t off = (size_t)1 << 20;
  float* buf0H = (float*)(ws + off); off += (size_t)512 * 128 * 128 * 4;  // 32MB
  float* buf0C = (float*)(ws + off); off += (size_t)512 * 128 * 128 * 4;
  float* buf1H = (float*)(ws + off); off += (size_t)256 * 128 * 128 * 4;  // 16MB
  float* buf1C = (float*)(ws + off); off += (size_t)256 * 128 * 128 * 4;
  float* hRootA = (float*)(ws + off); off += 128 * 128 * 4;
  float* hRootB = (float*)(ws + off); off += 128 * 128 * 4;
  float* cRootS = (float*)(ws + off); off += 128 * 128 * 4;

  prep_tables_kernel<<<18, 256, 0, stream>>>(emb, Wiou, biou, Wf, bf, Tiou, Tf);
  pack_weights_kernel<<<(64 * 40 * 32) / 256, 256, 0, stream>>>(Uiou, Uf, Wpack);

  float* bh[2] = {buf0H, buf1H};
  float* bc[2] = {buf0C, buf1C};
  const int* toks[2] = {xa, xb};
  float* roots[2] = {hRootA, hRootB};

  for (int tr = 0; tr < 2; ++tr) {
    const int* tok = toks[tr];
    // Level 9 (512 nodes): children are leaves, fused on the fly.
    tree_level_kernel<true, 4><<<(512 * 128) / 64, 256, 0, stream>>>(
        nullptr, nullptr, tok, 511, Wpack, Tiou, Tf, bh[0], bc[0]);
    int cur = 0;
    for (int lvl = 8; lvl >= 0; --lvl) {
      int M = 1 << lvl;
      int R = M * 128;
      int nxt = cur ^ 1;
      float* oH = (lvl == 0) ? roots[tr] : bh[nxt];
      float* oC = (lvl == 0) ? cRootS : bc[nxt];
      if (M >= 16) {
        tree_level_kernel<false, 4><<<R / 64, 256, 0, stream>>>(
            bh[cur], bc[cur], tok, M - 1, Wpack, Tiou, Tf, oH, oC);
      } else {
        tree_level_kernel<false, 1><<<R / 16, 256, 0, stream>>>(
            bh[cur], bc[cur], tok, M - 1, Wpack, Tiou, Tf, oH, oC);
      }
      cur = nxt;
    }
  }
  combine_kernel<<<128, 128, 0, stream>>>(Amat, U1, U2, bout, hRootA, hRootB, out);
}